// HouseHolderFlow_1580547968164
// MI455X (gfx1250) — compile-verified
//
#include <hip/hip_runtime.h>

// Householder reflection, B=16384 rows, L=4096 cols, fp32.
// One workgroup (256 threads = 8 wave32) per row.
// Data path: HBM --global_load_async_to_lds_b128--> LDS --ds_load_b128--> VGPR,
// dot reductions in-wave + cross-wave via LDS, NT b128 stores to HBM.

#define ROW_LEN 4096
#define THREADS 256
#define CHUNKS 4   // float4 chunks per thread: 4 * 256 * 4 floats = 4096

typedef float v4f __attribute__((ext_vector_type(4)));

__global__ __launch_bounds__(THREADS) void householder_row_kernel(
    const float* __restrict__ v,
    const float* __restrict__ z,
    float* __restrict__ out)
{
    __shared__ v4f sv[ROW_LEN / 4];   // 16 KB
    __shared__ v4f sz[ROW_LEN / 4];   // 16 KB
    __shared__ float rvz[THREADS / 32];
    __shared__ float rvv[THREADS / 32];

    const int t = threadIdx.x;
    const long long rowBase = (long long)blockIdx.x * ROW_LEN;

    // --- Stage 1: async global -> LDS (GVS mode: SGPR 64-bit base + VGPR 32-bit offset)
    const unsigned long long vbase = (unsigned long long)(v + rowBase);
    const unsigned long long zbase = (unsigned long long)(z + rowBase);
#pragma unroll
    for (int k = 0; k < CHUNKS; ++k) {
        const int c = t + k * THREADS;            // float4 chunk index within the row
        const unsigned goff = (unsigned)c * 16u;  // byte offset in global row
        const unsigned ldsv = (unsigned)(unsigned long long)(void*)&sv[c];
        const unsigned ldsz = (unsigned)(unsigned long long)(void*)&sz[c];
        asm volatile("global_load_async_to_lds_b128 %0, %1, %2 offset:0"
                     :: "v"(ldsv), "v"(goff), "s"(vbase) : "memory");
        asm volatile("global_load_async_to_lds_b128 %0, %1, %2 offset:0"
                     :: "v"(ldsz), "v"(goff), "s"(zbase) : "memory");
    }
    asm volatile("s_wait_asynccnt 0x0" ::: "memory");
    __syncthreads();  // every wave waited on its own ASYNCcnt; barrier makes all LDS writes visible

    // --- Stage 2: LDS -> registers, partial dot products
    v4f rv[CHUNKS], rz[CHUNKS];
    float pvz = 0.0f, pvv = 0.0f;
#pragma unroll
    for (int k = 0; k < CHUNKS; ++k) {
        const int c = t + k * THREADS;
        const v4f a = sv[c];
        const v4f b = sz[c];
        rv[k] = a; rz[k] = b;
        pvz = fmaf(a.x, b.x, fmaf(a.y, b.y, fmaf(a.z, b.z, fmaf(a.w, b.w, pvz))));
        pvv = fmaf(a.x, a.x, fmaf(a.y, a.y, fmaf(a.z, a.z, fmaf(a.w, a.w, pvv))));
    }

    // --- Stage 3: wave32 butterfly reduction, then cross-wave via LDS
#pragma unroll
    for (int off = 16; off > 0; off >>= 1) {
        pvz += __shfl_xor(pvz, off, 32);
        pvv += __shfl_xor(pvv, off, 32);
    }
    const int wave = t >> 5;
    if ((t & 31) == 0) { rvz[wave] = pvz; rvv[wave] = pvv; }
    __syncthreads();

    float svz = 0.0f, svv = 0.0f;
#pragma unroll
    for (int w = 0; w < THREADS / 32; ++w) { svz += rvz[w]; svv += rvv[w]; }
    const float scale = 2.0f * (svz / svv);

    // --- Stage 4: epilogue from registers, non-temporal b128 stores
    v4f* __restrict__ orow = (v4f*)(out + rowBase);
#pragma unroll
    for (int k = 0; k < CHUNKS; ++k) {
        const int c = t + k * THREADS;
        const v4f a = rv[k];
        const v4f b = rz[k];
        v4f o;
        o.x = fmaf(-scale, a.x, b.x);
        o.y = fmaf(-scale, a.y, b.y);
        o.z = fmaf(-scale, a.z, b.z);
        o.w = fmaf(-scale, a.w, b.w);
        __builtin_nontemporal_store(o, orow + c);
    }
}

extern "C" void kernel_launch(void* const* d_in, const int* in_sizes, int n_in,
                              void* d_out, int out_size, void* d_ws, size_t ws_size,
                              hipStream_t stream) {
    const float* v = (const float*)d_in[0];
    const float* z = (const float*)d_in[1];
    float* out = (float*)d_out;

    const int B = in_sizes[0] / ROW_LEN;  // 16384 rows
    householder_row_kernel<<<B, THREADS, 0, stream>>>(v, z, out);
}